// HetNetGNN_combine_50044958933536
// MI455X (gfx1250) — compile-verified
//
#include <hip/hip_runtime.h>

#define N_UE   100000
#define N_AP   100
#define N_EDGE 4000000

typedef __attribute__((ext_vector_type(2))) float v2f;
typedef __attribute__((ext_vector_type(8))) float v8f;

// ---------------------------------------------------------------------------
// Half-wave lane combines for converting WMMA C-layout -> B-layout (wave32).
// C layout of H (16x16, reg r): lanes0-15 = row r, lanes16-31 = row r+8.
// B layout for 4x16 K-chunk:    v0 lanes0-15 = K0, lanes16-31 = K2 (v1: K1/K3).
// ---------------------------------------------------------------------------
__device__ __forceinline__ float sel_lo_lo(float xlo, float ylo, int lane) {
  // result lanes0-15 = xlo(lane); lanes16-31 = ylo(lane-16)
  float ys = __shfl(ylo, lane & 15, 32);
  return (lane < 16) ? xlo : ys;
}
__device__ __forceinline__ float sel_hi_hi(float xhi, float yhi, int lane) {
  // result lanes0-15 = xhi(lane+16); lanes16-31 = yhi(lane)
  float xs = __shfl(xhi, (lane & 15) + 16, 32);
  return (lane < 16) ? xs : yhi;
}

// ---------------------------------------------------------------------------
// 16-node MLP tile: out^T = relu(W2^T (relu(W1^T X^T + b1)) + b2)
// X is 16 nodes x 2 features (K padded to 4). Uses V_WMMA_F32_16X16X4_F32.
// All weight/bias loads are unconditional (indices clamped in-bounds) followed
// by v_cndmask selects -- avoids exec-mask save/branch per load.
// Outputs valid at lanes 0-15: out0 = row0, out1 = row1 of the result.
// ---------------------------------------------------------------------------
template <int DOUT>
__device__ __forceinline__ void mlp16_wmma(
    float bx0, float bx1,
    const float* __restrict__ W1, const float* __restrict__ b1,
    const float* __restrict__ W2, const float* __restrict__ b2,
    int lane, float& out0, float& out1)
{
  const int  m  = lane & 15;
  const bool lo = lane < 16;

  // A1 = W1^T (16 hidden x K=4; K=2,3 zero pad). A layout: v0=K0|K2, v1=K1|K3.
  // W1 has 2*16 entries; W1[m], W1[16+m] are in-bounds for every lane.
  const float w1a = W1[m];
  const float w1b = W1[16 + m];
  v2f A1;
  A1.x = lo ? w1a : 0.0f;
  A1.y = lo ? w1b : 0.0f;
  v2f B1;  B1.x = bx0;  B1.y = bx1;

  v8f C;
#pragma unroll
  for (int r = 0; r < 8; ++r) C[r] = b1[lo ? r : r + 8];   // lane-varying addr, unconditional

  v8f H = __builtin_amdgcn_wmma_f32_16x16x4_f32(false, A1, false, B1,
                                                (short)0, C, false, false);
#pragma unroll
  for (int r = 0; r < 8; ++r) H[r] = fmaxf(H[r], 0.0f);

  // GEMM2: D = W2^T (16 x K=16) * H^T, 4 chained K=4 WMMAs.
  const float bb0 = b2[0];
  float bb1v = 0.0f;
  if (DOUT > 1) bb1v = b2[1];

  v8f acc;
#pragma unroll
  for (int r = 0; r < 8; ++r) acc[r] = 0.0f;
  acc[0] = lo ? bb0 : 0.0f;
  if (DOUT > 1) acc[1] = lo ? bb1v : 0.0f;

  const int  mc  = (m < DOUT) ? m : (DOUT - 1);   // clamp: loads stay in-bounds
  const bool mok = (m < DOUT);
#pragma unroll
  for (int kb = 0; kb < 4; ++kb) {
    const int kA = 4 * kb + (lo ? 0 : 2);
    const int kB = 4 * kb + (lo ? 1 : 3);
    const float wA = W2[kA * DOUT + mc];   // max idx = 16*DOUT-1: in-bounds
    const float wB = W2[kB * DOUT + mc];
    v2f A2;
    A2.x = mok ? wA : 0.0f;
    A2.y = mok ? wB : 0.0f;
    v2f B2;
    if      (kb == 0) { B2.x = sel_lo_lo(H[0], H[2], lane); B2.y = sel_lo_lo(H[1], H[3], lane); }
    else if (kb == 1) { B2.x = sel_lo_lo(H[4], H[6], lane); B2.y = sel_lo_lo(H[5], H[7], lane); }
    else if (kb == 2) { B2.x = sel_hi_hi(H[0], H[2], lane); B2.y = sel_hi_hi(H[1], H[3], lane); }
    else              { B2.x = sel_hi_hi(H[4], H[6], lane); B2.y = sel_hi_hi(H[5], H[7], lane); }
    acc = __builtin_amdgcn_wmma_f32_16x16x4_f32(false, A2, false, B2,
                                                (short)0, acc, false, false);
  }
  out0 = fmaxf(acc[0], 0.0f);
  out1 = fmaxf(acc[1], 0.0f);
}

// ---------------------------------------------------------------------------
__global__ void fill_zero_kernel(float* __restrict__ p, int n) {
  const int stride = gridDim.x * blockDim.x;
  for (int i = blockIdx.x * blockDim.x + threadIdx.x; i < n; i += stride) p[i] = 0.0f;
}

// Segment counts (layer-invariant): UE via global atomics (100K bins),
// AP via LDS accumulation (100 bins, heavy contention otherwise).
__global__ void count_kernel(const int* __restrict__ dst_d, const int* __restrict__ dst_u,
                             float* __restrict__ cnt_ue, float* __restrict__ cnt_ap,
                             int n_edges) {
  __shared__ float lcnt[N_AP];
  for (int i = threadIdx.x; i < N_AP; i += blockDim.x) lcnt[i] = 0.0f;
  __syncthreads();
  const int stride = gridDim.x * blockDim.x;
  for (int e = blockIdx.x * blockDim.x + threadIdx.x; e < n_edges; e += stride) {
    atomicAdd(&cnt_ue[dst_d[e]], 1.0f);
    atomicAdd(&lcnt[dst_u[e]], 1.0f);
  }
  __syncthreads();
  for (int i = threadIdx.x; i < N_AP; i += blockDim.x)
    atomicAdd(&cnt_ap[i], lcnt[i]);
}

// Downlink: per-edge MLP (1->16->1, relu after both) then scatter-add to UE.
__global__ void edge_down_kernel(const float* __restrict__ ea, const int* __restrict__ dst,
                                 const float* __restrict__ W1, const float* __restrict__ b1,
                                 const float* __restrict__ W2, const float* __restrict__ b2,
                                 float* __restrict__ sum_ue, int n_edges) {
  float w1[16], bb1[16], w2[16];
#pragma unroll
  for (int j = 0; j < 16; ++j) { w1[j] = W1[j]; bb1[j] = b1[j]; w2[j] = W2[j]; }
  const float bb2 = b2[0];
  const int stride = gridDim.x * blockDim.x;
  for (int e = blockIdx.x * blockDim.x + threadIdx.x; e < n_edges; e += stride) {
    __builtin_prefetch(&ea[2 * (e + stride)], 0, 1);
    const float a = ea[2 * e];
    float acc = bb2;
#pragma unroll
    for (int j = 0; j < 16; ++j) {
      const float h = fmaxf(fmaf(a, w1[j], bb1[j]), 0.0f);
      acc = fmaf(h, w2[j], acc);
    }
    atomicAdd(&sum_ue[dst[e]], fmaxf(acc, 0.0f));
  }
}

// UE node update: aggr = seg_mean(downlink), res = upd_ue MLP, nm = msg_ue MLP.
// One wave per 16 nodes; both 2->16->{1,2} MLPs run on the fp32 WMMA path.
// Loads are clamped+unconditional; only the final store block is predicated.
__global__ void ue_update_kernel(const float* __restrict__ x_in,
                                 const float* __restrict__ sum_ue,
                                 const float* __restrict__ cnt_ue,
                                 const float* __restrict__ Wu1, const float* __restrict__ bu1,
                                 const float* __restrict__ Wu2, const float* __restrict__ bu2,
                                 const float* __restrict__ Wm1, const float* __restrict__ bm1,
                                 const float* __restrict__ Wm2, const float* __restrict__ bm2,
                                 float* __restrict__ x_out, float* __restrict__ nm_out,
                                 int n_nodes) {
  const int  lane = threadIdx.x & 31;
  const int  wave = (blockIdx.x * blockDim.x + threadIdx.x) >> 5;
  const int  base = wave * 16;
  if (base >= n_nodes) return;  // wave-uniform exit (EXEC stays full for WMMA)
  const int  m      = lane & 15;
  const bool lo     = lane < 16;
  const int  node   = base + m;
  const int  node_c = (node < n_nodes) ? node : (n_nodes - 1);  // clamp for loads
  const bool valid  = lo && (node < n_nodes);

  const float x0 = x_in[2 * node_c];
  const float x1 = x_in[2 * node_c + 1];
  const float bx0 = valid ? x0 : 0.0f;   // lanes>=16 and OOB -> 0 (K pad)
  const float bx1 = valid ? x1 : 0.0f;

  float r0, r1, n0, n1;
  mlp16_wmma<1>(bx0, bx1, Wu1, bu1, Wu2, bu2, lane, r0, r1);
  mlp16_wmma<2>(bx0, bx1, Wm1, bm1, Wm2, bm2, lane, n0, n1);
  (void)r1;

  const float aggr = sum_ue[node_c] / fmaxf(cnt_ue[node_c], 1.0f);
  if (valid) {
    x_out[2 * node]      = x0;          // col 0 carried through unchanged
    x_out[2 * node + 1]  = aggr + r0;
    nm_out[2 * node]     = n0;
    nm_out[2 * node + 1] = n1;
  }
}

// Uplink: per-edge MLP (1->16->2) + gather nm[src], LDS-accumulated scatter to AP.
__global__ void edge_up_kernel(const float* __restrict__ ea,
                               const int* __restrict__ src, const int* __restrict__ dst,
                               const float* __restrict__ nm,
                               const float* __restrict__ W1, const float* __restrict__ b1,
                               const float* __restrict__ W2, const float* __restrict__ b2,
                               float* __restrict__ sum_ap, int n_edges) {
  __shared__ float lsum[2 * N_AP];
  for (int i = threadIdx.x; i < 2 * N_AP; i += blockDim.x) lsum[i] = 0.0f;
  __syncthreads();
  float w1[16], bb1[16], w20[16], w21[16];
#pragma unroll
  for (int j = 0; j < 16; ++j) {
    w1[j] = W1[j]; bb1[j] = b1[j];
    w20[j] = W2[2 * j]; w21[j] = W2[2 * j + 1];
  }
  const float bb20 = b2[0], bb21 = b2[1];
  const int stride = gridDim.x * blockDim.x;
  for (int e = blockIdx.x * blockDim.x + threadIdx.x; e < n_edges; e += stride) {
    __builtin_prefetch(&ea[2 * (e + stride)], 0, 1);
    const float a = ea[2 * e];
    float a0 = bb20, a1 = bb21;
#pragma unroll
    for (int j = 0; j < 16; ++j) {
      const float h = fmaxf(fmaf(a, w1[j], bb1[j]), 0.0f);
      a0 = fmaf(h, w20[j], a0);
      a1 = fmaf(h, w21[j], a1);
    }
    a0 = fmaxf(a0, 0.0f);
    a1 = fmaxf(a1, 0.0f);
    const int s = src[e];
    const int d = dst[e];
    atomicAdd(&lsum[2 * d],     nm[2 * s]     + a0);
    atomicAdd(&lsum[2 * d + 1], nm[2 * s + 1] + a1);
  }
  __syncthreads();
  for (int i = threadIdx.x; i < 2 * N_AP; i += blockDim.x)
    atomicAdd(&sum_ap[i], lsum[i]);
}

__global__ void ap_finalize_kernel(const float* __restrict__ sum_ap,
                                   const float* __restrict__ cnt_ap,
                                   float* __restrict__ out) {
  const int i = blockIdx.x * blockDim.x + threadIdx.x;
  if (i < N_AP) {
    const float c = fmaxf(cnt_ap[i], 1.0f);
    out[2 * i]     = sum_ap[2 * i]     / c;
    out[2 * i + 1] = sum_ap[2 * i + 1] / c;
  }
}

// ---------------------------------------------------------------------------
extern "C" void kernel_launch(void* const* d_in, const int* in_sizes, int n_in,
                              void* d_out, int out_size, void* d_ws, size_t ws_size,
                              hipStream_t stream) {
  const float* x_ue  = (const float*)d_in[0];
  const float* ea_d  = (const float*)d_in[2];
  const float* ea_u  = (const float*)d_in[3];
  // d_in[4] (src_d) is dead: downlink messages ignore x_j.
  const int*   dst_d = (const int*)d_in[5];
  const int*   src_u = (const int*)d_in[6];
  const int*   dst_u = (const int*)d_in[7];

  // params flattened layer-major, mlp key order {edge_d, edge_u, msg_ue, upd_ue},
  // tuple order (W1, b1, W2, b2).
  auto P = [&](int layer, int mlp, int t) -> const float* {
    return (const float*)d_in[8 + layer * 16 + mlp * 4 + t];
  };

  float* ws     = (float*)d_ws;
  float* sum_ue = ws;                       // N_UE
  float* cnt_ue = ws + N_UE;                // N_UE
  float* x_mid  = ws + 2 * N_UE;            // N_UE*2
  float* nm_buf = ws + 4 * N_UE;            // N_UE*2
  float* sum_ap = ws + 6 * N_UE;            // 2*N_AP
  float* cnt_ap = ws + 6 * N_UE + 2 * N_AP; // N_AP
  float* out_ue = (float*)d_out;            // N_UE*2
  float* out_ap = (float*)d_out + 2 * N_UE; // N_AP*2

  const dim3 blk(256);
  const int  ue_blocks = (((N_UE + 15) / 16) * 32 + 255) / 256;

  fill_zero_kernel<<<dim3((N_UE + 255) / 256), blk, 0, stream>>>(sum_ue, N_UE);
  fill_zero_kernel<<<dim3((N_UE + 255) / 256), blk, 0, stream>>>(cnt_ue, N_UE);
  fill_zero_kernel<<<dim3(1), blk, 0, stream>>>(sum_ap, 3 * N_AP); // sum_ap + cnt_ap
  count_kernel<<<dim3(2048), blk, 0, stream>>>(dst_d, dst_u, cnt_ue, cnt_ap, N_EDGE);

  // ---- Layer 0 (uplink output x_ap discarded -> downlink path only) ----
  edge_down_kernel<<<dim3(4096), blk, 0, stream>>>(
      ea_d, dst_d, P(0,0,0), P(0,0,1), P(0,0,2), P(0,0,3), sum_ue, N_EDGE);
  ue_update_kernel<<<dim3(ue_blocks), blk, 0, stream>>>(
      x_ue, sum_ue, cnt_ue,
      P(0,3,0), P(0,3,1), P(0,3,2), P(0,3,3),   // upd_ue
      P(0,2,0), P(0,2,1), P(0,2,2), P(0,2,3),   // msg_ue
      x_mid, nm_buf, N_UE);

  // ---- Layer 1 (full) ----
  fill_zero_kernel<<<dim3((N_UE + 255) / 256), blk, 0, stream>>>(sum_ue, N_UE);
  edge_down_kernel<<<dim3(4096), blk, 0, stream>>>(
      ea_d, dst_d, P(1,0,0), P(1,0,1), P(1,0,2), P(1,0,3), sum_ue, N_EDGE);
  ue_update_kernel<<<dim3(ue_blocks), blk, 0, stream>>>(
      x_mid, sum_ue, cnt_ue,
      P(1,3,0), P(1,3,1), P(1,3,2), P(1,3,3),
      P(1,2,0), P(1,2,1), P(1,2,2), P(1,2,3),
      out_ue, nm_buf, N_UE);
  edge_up_kernel<<<dim3(2048), blk, 0, stream>>>(
      ea_u, src_u, dst_u, nm_buf,
      P(1,1,0), P(1,1,1), P(1,1,2), P(1,1,3), sum_ap, N_EDGE);
  ap_finalize_kernel<<<dim3(1), dim3(128), 0, stream>>>(sum_ap, cnt_ap, out_ap);
}